// YoloStrategy_27590869909654
// MI455X (gfx1250) — compile-verified
//
#include <hip/hip_runtime.h>
#include <hip/hip_bf16.h>

// ---------------------------------------------------------------------------
// YOLO decode + greedy NMS for MI455X (gfx1250).
//  Kernel 1: wave32-per-box decode (box corners, score, class).
//  Kernel 2: single persistent workgroup (32 wave32 waves). Scores (f32) +
//    classes (u8) resident in LDS, staged via gfx1250 async global->LDS B128
//    (ASYNCcnt / s_wait_asynccnt). Each of the 300 sequential NMS rounds uses
//    only TWO workgroup barriers: packed-u64-key wave reduction + ds_max_u64,
//    then on-the-fly IoU suppression (self-kill via IoU(self)=1).
// ---------------------------------------------------------------------------

#define NBOX 10647
#define NPAD 10656          // NBOX rounded up to multiple of 32 (and of 4)
#define NVEC (NPAD / 4)     // 2664 float4 score vectors
#define NCLSV (NPAD / 16)   // 666 16-byte class chunks
#define NCLS 80
#define FEAT 85
#define MAX_DET 300
#define SCORE_THRESH 0.3f
#define IOU_THRESH 0.45f
#define NMS_THREADS 1024

// ---------------------------------------------------------------------------
// Decode: one wave (32 lanes) per box.
// ---------------------------------------------------------------------------
__global__ __launch_bounds__(512)
void yolo_decode_kernel(const float* __restrict__ pred,
                        const int* __restrict__ org_h_p,
                        const int* __restrict__ org_w_p,
                        float* __restrict__ coor,          // NPAD * 4 floats
                        float* __restrict__ scores,        // NPAD floats
                        unsigned char* __restrict__ cls)   // NPAD bytes
{
    const int gtid = blockIdx.x * blockDim.x + threadIdx.x;
    const int box  = gtid >> 5;
    const int lane = gtid & 31;
    if (box >= NPAD) return;
    if (box >= NBOX) {                      // padding boxes: dead entries
        if (lane == 0) scores[box] = 0.0f;
        return;
    }

    const float* row = pred + (size_t)box * FEAT;

    // ---- per-wave argmax over 80 class probs (first-occurrence ties) ----
    float bv = -1.0f;
    int   bc = 0x7fffffff;
    for (int c = lane; c < NCLS; c += 32) {
        float p = row[5 + c];
        if (p > bv) { bv = p; bc = c; }     // per-lane candidates are increasing
    }
    for (int off = 16; off > 0; off >>= 1) {
        float ov = __shfl_down(bv, (unsigned)off, 32);
        int   oc = __shfl_down(bc, (unsigned)off, 32);
        if (ov > bv || (ov == bv && oc < bc)) { bv = ov; bc = oc; }
    }

    if (lane == 0) {
        const float x = row[0], y = row[1], w = row[2], h = row[3], conf = row[4];
        const int oh = *org_h_p, ow = *org_w_p;

        // ratio/shift in f64 (Python scalars), applied in f32 (JAX weak type)
        const double rr = fmin(416.0 / (double)ow, 416.0 / (double)oh);
        const float  dw = (float)((416.0 - rr * (double)ow) * 0.5);
        const float  dh = (float)((416.0 - rr * (double)oh) * 0.5);
        const float  r  = (float)rr;

        float x1 = (x - w * 0.5f - dw) / r;
        float y1 = (y - h * 0.5f - dh) / r;
        float x2 = (x + w * 0.5f - dw) / r;
        float y2 = (y + h * 0.5f - dh) / r;

        x1 = fmaxf(x1, 0.0f);
        y1 = fmaxf(y1, 0.0f);
        x2 = fminf(x2, (float)(ow - 1));
        y2 = fminf(y2, (float)(oh - 1));

        const bool invalid = (x1 > x2) || (y1 > y2);
        if (invalid) { x1 = y1 = x2 = y2 = 0.0f; }

        const float area       = (x2 - x1) * (y2 - y1);
        const bool  scale_mask = sqrtf(area) > 0.0f;

        const float sc   = conf * bv;
        const bool  keep = scale_mask && (sc > SCORE_THRESH);

        scores[box] = keep ? sc : 0.0f;
        cls[box]    = (unsigned char)bc;
        float4 cc; cc.x = x1; cc.y = y1; cc.z = x2; cc.w = y2;
        ((float4*)coor)[box] = cc;
    }
}

// packed argmax key: (score_bits << 32) | (0xFFFFFFFF - idx)
// score >= 0 -> f32 bit pattern is monotone; larger key == higher score,
// ties broken toward SMALLER index (jnp.argmax first-occurrence).
__device__ __forceinline__ unsigned long long pack_key(float v, int idx) {
    return ((unsigned long long)__float_as_uint(v) << 32) |
           (unsigned long long)(0xFFFFFFFFu - (unsigned)idx);
}

// ---------------------------------------------------------------------------
// NMS: single workgroup of 1024 threads (32 wave32 waves), 2 barriers/round.
// ---------------------------------------------------------------------------
__global__ __launch_bounds__(NMS_THREADS)
void yolo_nms_kernel(const float4* __restrict__ coor4,
                     const float* __restrict__ scores_g,
                     const unsigned char* __restrict__ cls_g,
                     float* __restrict__ out)
{
    __shared__ __align__(16) float              s_sc[NPAD];   // 42624 B
    __shared__ __align__(16) unsigned char      s_cls[NPAD];  // 10656 B
    __shared__ unsigned long long               s_key[2];     // parity-buffered

    const int tid = threadIdx.x;
    const int NT  = NMS_THREADS;

    // LDS byte offsets: low 32 bits of the generic pointer are the LDS offset
    // (ISA 10.2: LDS aperture keeps offset in addr[31:0]).
    const unsigned sc_base  = (unsigned)(size_t)&s_sc[0];
    const unsigned cls_base = (unsigned)(size_t)&s_cls[0];

    // ---- async stage: scores, 16 B per op (NVEC float4 chunks) ----
    for (int j = tid; j < NVEC; j += NT) {
        unsigned           dst = sc_base + (unsigned)j * 16u;
        unsigned long long src = (unsigned long long)(const void*)(scores_g + 4 * j);
        asm volatile("global_load_async_to_lds_b128 %0, %1, off"
                     :: "v"(dst), "v"(src) : "memory");
    }
    // ---- async stage: classes, 16 B per op ----
    for (int j = tid; j < NCLSV; j += NT) {
        unsigned           dst = cls_base + (unsigned)j * 16u;
        unsigned long long src = (unsigned long long)(const void*)(cls_g + 16 * j);
        asm volatile("global_load_async_to_lds_b128 %0, %1, off"
                     :: "v"(dst), "v"(src) : "memory");
    }
    asm volatile("s_wait_asynccnt 0" ::: "memory");

    if (tid == 0) { s_key[0] = 0ull; s_key[1] = 0ull; }
    // padded score tail is 0 in global (decode writes it), staged as-is.
    __syncthreads();

    const float4* s_sc4 = (const float4*)s_sc;

    for (int step = 0; step < MAX_DET; ++step) {
        const int p = step & 1;

        // ---------- block argmax: vectorized scan + wave reduce + ds_max_u64
        unsigned long long best = 0ull;
        for (int j = tid; j < NVEC; j += NT) {
            float4 v = s_sc4[j];
            const int i0 = 4 * j;
            unsigned long long k;
            k = pack_key(v.x, i0 + 0); if (k > best) best = k;
            k = pack_key(v.y, i0 + 1); if (k > best) best = k;
            k = pack_key(v.z, i0 + 2); if (k > best) best = k;
            k = pack_key(v.w, i0 + 3); if (k > best) best = k;
        }
        for (int off = 16; off > 0; off >>= 1) {
            unsigned long long o = __shfl_down(best, (unsigned)off, 32);
            if (o > best) best = o;
        }
        if ((tid & 31) == 0) {
            atomicMax(&s_key[p], best);          // ds_max_u64
        }
        __syncthreads();                          // barrier #1

        const unsigned long long kk = s_key[p];
        const float bv = __uint_as_float((unsigned)(kk >> 32));
        const int   bi = (int)(0xFFFFFFFFu - (unsigned)(kk & 0xFFFFFFFFu));
        const int   valid = (bv > 0.0f) ? 1 : 0;

        if (tid == 0) {
            float4 bb; bb.x = bb.y = bb.z = bb.w = 0.0f;
            float  bcf = 0.0f;
            if (valid) { bb = coor4[bi]; bcf = (float)(int)s_cls[bi]; }
            float* orow = out + step * 6;
            orow[0] = bb.x; orow[1] = bb.y; orow[2] = bb.z; orow[3] = bb.w;
            orow[4] = valid ? bv : 0.0f;
            orow[5] = bcf;
            s_key[1 - p] = 0ull;                  // reset other parity slot
        }

        // ---------- on-the-fly IoU suppression (overlaps row emit) ----------
        if (valid) {
            const float4 pb  = coor4[bi];         // broadcast load (same addr)
            const int    pc  = (int)s_cls[bi];
            const float parea = (pb.z - pb.x) * (pb.w - pb.y);
            for (int j = tid; j < NVEC; j += NT) {
                float4 sv = s_sc4[j];
                const int i0 = 4 * j;
                #pragma unroll
                for (int k = 0; k < 4; ++k) {
                    const float s = (k == 0) ? sv.x : (k == 1) ? sv.y
                                  : (k == 2) ? sv.z : sv.w;
                    const int i = i0 + k;
                    if (s > 0.0f && (int)s_cls[i] == pc) {
                        float4 c = coor4[i];
                        float iw = fminf(pb.z, c.z) - fmaxf(pb.x, c.x);
                        float ih = fminf(pb.w, c.w) - fmaxf(pb.y, c.y);
                        iw = fmaxf(iw, 0.0f);
                        ih = fmaxf(ih, 0.0f);
                        const float inter = iw * ih;
                        const float carea = (c.z - c.x) * (c.w - c.y);
                        const float uni   = parea + carea - inter;
                        const float iou   = inter / uni;  // 0/0->NaN->no kill
                        if (iou > IOU_THRESH || i == bi) s_sc[i] = 0.0f;
                    }
                }
            }
        }
        __syncthreads();                          // barrier #2
    }
}

// ---------------------------------------------------------------------------
extern "C" void kernel_launch(void* const* d_in, const int* in_sizes, int n_in,
                              void* d_out, int out_size, void* d_ws, size_t ws_size,
                              hipStream_t stream)
{
    (void)in_sizes; (void)n_in; (void)out_size; (void)ws_size;

    const float* pred  = (const float*)d_in[0];
    const int*   org_h = (const int*)d_in[1];
    const int*   org_w = (const int*)d_in[2];
    float*       out   = (float*)d_out;

    char* ws = (char*)d_ws;
    float*         coor   = (float*)ws;                              // NPAD*16 B
    float*         scores = (float*)(ws + (size_t)NPAD * 16);        // NPAD*4 B
    unsigned char* cls    = (unsigned char*)(ws + (size_t)NPAD * 16
                                                + (size_t)NPAD * 4); // NPAD B

    // 666 blocks * 512 threads = 10656 waves = NPAD boxes (wave per box)
    yolo_decode_kernel<<<dim3((NPAD * 32) / 512), dim3(512), 0, stream>>>(
        pred, org_h, org_w, coor, scores, cls);

    yolo_nms_kernel<<<dim3(1), dim3(NMS_THREADS), 0, stream>>>(
        (const float4*)coor, scores, cls, out);
}